// JeffressLinear_49641232007669
// MI455X (gfx1250) — compile-verified
//
#include <hip/hip_runtime.h>
#include <hip/hip_bf16.h>

// Problem dims (fixed by the reference)
#define T_DIM 128
#define N_DIM 32
#define C_DIM 64
#define D_DIM 64

typedef float v4f __attribute__((ext_vector_type(4)));

// One block per (n,c). 256 threads = 8 wave32.
__global__ __launch_bounds__(256) void jeffress_kernel(
    const float* __restrict__ x,          // (T,N,C,2)
    const float* __restrict__ noise,      // (N,C,D,2)
    const float* __restrict__ log_delay,  // (D,1)
    const float* __restrict__ log_weight, // scalar
    float* __restrict__ out)              // (T,N,C,D)
{
    __shared__ __align__(16) float xs[T_DIM * 2];   // x[t][j] slice for this (n,c)
    __shared__ int rd_lds[2 * D_DIM];               // rd[j][d]
    __shared__ unsigned long long wmax[8];          // per-wave argmax partials
    __shared__ int am_lds[2];                       // argmax_t per j

    const int tid = threadIdx.x;
    const int nc  = blockIdx.x;          // n*C + c
    const int xoff = nc * 2;             // float offset of x[0,n,c,0]

    // ---- Stage x[:, n, c, :] (128 rows x 8B, row stride N*C*2 floats = 16KB)
    //      into LDS using gfx1250 async global->LDS DMA (ASYNCcnt path). ----
    if (tid < T_DIM) {
        const float* gp = x + (size_t)tid * (N_DIM * C_DIM * 2) + xoff;
        unsigned lds_addr = (unsigned)(uintptr_t)(&xs[tid * 2]); // low 32b = LDS byte offset
        asm volatile("global_load_async_to_lds_b64 %0, %1, off"
                     :: "v"(lds_addr), "v"(gp)
                     : "memory");
    }
    asm volatile("s_wait_asynccnt 0x0" ::: "memory");
    __syncthreads();

    // ---- First-occurrence argmax over t, for j=0 (threads 0..127) and
    //      j=1 (threads 128..255). x is uniform [0,1) so non-negative f32
    //      bit patterns order monotonically: key = (bits(v)<<8) | (127-t),
    //      max-key => max value, ties -> smallest t. ----
    {
        const int jj = tid >> 7;         // 0 or 1
        const int tt = tid & 127;        // time index this thread owns
        const float v = xs[tt * 2 + jj];
        unsigned long long key =
            ((unsigned long long)__float_as_uint(v) << 8) | (unsigned)(127 - tt);
        #pragma unroll
        for (int off = 16; off > 0; off >>= 1) {
            unsigned long long o = __shfl_xor(key, off, 32);   // wave32 shuffle
            key = (key > o) ? key : o;
        }
        if ((tid & 31) == 0) wmax[tid >> 5] = key;
    }
    __syncthreads();
    if (tid < 2) {
        unsigned long long k = wmax[tid * 4];
        #pragma unroll
        for (int i = 1; i < 4; ++i) {
            unsigned long long o = wmax[tid * 4 + i];
            k = (k > o) ? k : o;
        }
        am_lds[tid] = 127 - (int)(k & 0xffu);
    }
    __syncthreads();

    // ---- Stochastic round + clamp -> rd[j][d] ----
    if (tid < 2 * D_DIM) {
        const int j = tid >> 6;
        const int d = tid & 63;
        const float ld    = log_delay[(j == 0) ? d : (D_DIM - 1 - d)]; // flip for j=1
        const float delay = expf(ld);
        const float fl    = floorf(delay);
        const float frac  = delay - fl;
        const float nv    = noise[(size_t)nc * (D_DIM * 2) + d * 2 + j];
        int r = (int)fl + ((nv < frac) ? 1 : 0);
        const int cap = (T_DIM - 1) - am_lds[j];
        r = (r < cap) ? r : cap;
        rd_lds[tid] = r;                 // tid == j*64 + d
    }
    __syncthreads();

    // ---- Circular gather + weighted sum over j, streamed out with wide
    //      non-temporal B128 stores. Thread owns 4 consecutive d; half-wave
    //      (16 lanes) writes one full 256B d-row per t. ----
    const float w  = expf(log_weight[0]);
    const int   d4 = (tid & 15) * 4;                 // base d of this thread
    const int   tq = tid >> 4;                       // 0..15 -> 8 t's each
    int r0[4], r1[4];
    #pragma unroll
    for (int i = 0; i < 4; ++i) {
        r0[i] = rd_lds[d4 + i];
        r1[i] = rd_lds[64 + d4 + i];
    }
    const size_t obase = (size_t)nc * D_DIM + d4;
    #pragma unroll
    for (int k = 0; k < 8; ++k) {
        const int t = tq * 8 + k;
        v4f v;
        #pragma unroll
        for (int i = 0; i < 4; ++i) {
            const int s0 = (t - r0[i]) & (T_DIM - 1);    // mod 128, non-negative
            const int s1 = (t - r1[i]) & (T_DIM - 1);
            v[i] = w * (xs[s0 * 2] + xs[s1 * 2 + 1]);
        }
        __builtin_nontemporal_store(
            v, (v4f*)(out + (size_t)t * (N_DIM * C_DIM * D_DIM) + obase));
    }
}

extern "C" void kernel_launch(void* const* d_in, const int* in_sizes, int n_in,
                              void* d_out, int out_size, void* d_ws, size_t ws_size,
                              hipStream_t stream) {
    const float* x          = (const float*)d_in[0];
    const float* noise      = (const float*)d_in[1];
    const float* log_delay  = (const float*)d_in[2];
    const float* log_weight = (const float*)d_in[3];
    float* out = (float*)d_out;
    (void)in_sizes; (void)n_in; (void)out_size; (void)d_ws; (void)ws_size;

    jeffress_kernel<<<dim3(N_DIM * C_DIM), dim3(256), 0, stream>>>(
        x, noise, log_delay, log_weight, out);
}